// ContinuousFilterConv_61400852463921
// MI455X (gfx1250) — compile-verified
//
#include <hip/hip_runtime.h>
#include <hip/hip_bf16.h>

typedef __attribute__((ext_vector_type(2))) float v2f;
typedef __attribute__((ext_vector_type(8))) float v8f;

#define NNODES 50000
#define NEDGES 800000
#define HID 64
#define NTILES (NEDGES / 16)   /* 50000 16-edge tiles */
#define HSTRIDE 68             /* padded LDS row stride (floats) -> conflict-free */
#define WPB 4                  /* waves per block (blockDim = 128) */

// ---------------------------------------------------------------------------
// Prologue: zero the accumulation buffers (d_out = num, d_ws = den)
// ---------------------------------------------------------------------------
__global__ __launch_bounds__(256) void cfc_zero_kernel(float* __restrict__ num,
                                                       float* __restrict__ den) {
  int stride = gridDim.x * blockDim.x;
  for (int i = blockIdx.x * blockDim.x + threadIdx.x; i < NNODES * HID; i += stride)
    num[i] = 0.0f;
  for (int i = blockIdx.x * blockDim.x + threadIdx.x; i < NNODES; i += stride)
    den[i] = 0.0f;
}

// ---------------------------------------------------------------------------
// Fused edge kernel: MLP (2x WMMA GEMM) + tanh + gather + attention + scatter
// One wave32 processes a 16-edge tile: D = 16x64, K = 64 (16 x WMMA f32 16x16x4)
// ---------------------------------------------------------------------------
__global__ __launch_bounds__(WPB * 32) void cfc_edge_kernel(
    const float* __restrict__ atom,   // [NNODES,64]
    const float* __restrict__ eemb,   // [NEDGES,64]
    const float* __restrict__ W1,     // [64,64]
    const float* __restrict__ b1,     // [64]
    const float* __restrict__ W2,     // [64,64]
    const float* __restrict__ b2,     // [64]
    const float* __restrict__ coef,   // [64]
    const int*   __restrict__ elist,  // [NEDGES,2]  (dest, src)
    float* __restrict__ num,          // [NNODES,64] pre-zeroed
    float* __restrict__ den)          // [NNODES]    pre-zeroed
{
  // W1/W2 pre-swizzled into the WMMA B-operand layout: [ntile][kchunk][lane]
  __shared__ v2f   w1s[4][16][32];
  __shared__ v2f   w2s[4][16][32];
  __shared__ float hbuf[WPB][16 * HSTRIDE];  // per-wave D->A relayout staging
  __shared__ int   idxbuf[WPB][32];          // [0..15]=dest node, [16..31]=src node

  const int tid = threadIdx.x;

  // Cooperative one-time weight swizzle (B layout: VGPR0 = rows K,K+1 for
  // lanes 0-15 / rows K+2,K+3 for lanes 16-31, cols = lane&15 within N-tile).
  for (int f = tid; f < 4 * 16 * 32; f += blockDim.x) {
    int ln = f & 31, kc = (f >> 5) & 15, t = f >> 9;
    int h = ln >> 4, cc = ln & 15;
    int k0 = 4 * kc + 2 * h;
    v2f w;
    w.x = W1[(k0 + 0) * HID + 16 * t + cc];
    w.y = W1[(k0 + 1) * HID + 16 * t + cc];
    w1s[t][kc][ln] = w;
    w.x = W2[(k0 + 0) * HID + 16 * t + cc];
    w.y = W2[(k0 + 1) * HID + 16 * t + cc];
    w2s[t][kc][ln] = w;
  }
  __syncthreads();

  const int lane = tid & 31;
  const int wv   = tid >> 5;
  const int half = lane >> 4;   // which 16-lane half of the wave
  const int c    = lane & 15;   // column within an N-tile / row within M-tile

  float bias1[4], bias2[4], cf[4];
#pragma unroll
  for (int t = 0; t < 4; ++t) {
    bias1[t] = b1[16 * t + c];
    bias2[t] = b2[16 * t + c];
    cf[t]    = coef[16 * t + c];
  }

  float* hrow = hbuf[wv];
  int*   ib   = idxbuf[wv];

  const int gwave  = blockIdx.x * WPB + wv;
  const int nwaves = gridDim.x * WPB;

  for (int tile = gwave; tile < NTILES; tile += nwaves) {
    const int ebase = tile * 16;

    // ---- A operand: edge_emb tile, 16 edges x 64 features, WMMA A layout ----
    const float* arow = eemb + (size_t)(ebase + c) * HID + 2 * half;
    v2f areg[16];
#pragma unroll
    for (int kc = 0; kc < 16; ++kc)
      areg[kc] = *(const v2f*)(arow + 4 * kc);

    // Prefetch next tile's edge rows (global_prefetch_b8)
    if (tile + nwaves < NTILES)
      __builtin_prefetch(arow + (size_t)nwaves * 16 * HID, 0, 0);

    // ---- edge indices to LDS (per-wave; DS ops are in-order within a wave) --
    if (lane < 16) {
      ib[c]      = elist[2 * (ebase + c) + 0];  // dest
      ib[16 + c] = elist[2 * (ebase + c) + 1];  // src
    }

    // ---- GEMM1: H = tanh(X @ W1 + b1),  acc[t] covers N-cols 16t..16t+15 ----
    v8f acc[4];
#pragma unroll
    for (int t = 0; t < 4; ++t) {
      v8f z;
#pragma unroll
      for (int j = 0; j < 8; ++j) z[j] = bias1[t];
      acc[t] = z;
    }
#pragma unroll
    for (int kc = 0; kc < 16; ++kc) {
#pragma unroll
      for (int t = 0; t < 4; ++t) {
        acc[t] = __builtin_amdgcn_wmma_f32_16x16x4_f32(
            false, areg[kc], false, w1s[t][kc][lane], (short)0, acc[t],
            false, false);
      }
    }

    // ---- tanh + stage D-layout -> row-major 16x64 (padded) in LDS ----------
#pragma unroll
    for (int t = 0; t < 4; ++t) {
#pragma unroll
      for (int j = 0; j < 8; ++j) {
        int row = j + 8 * half;  // C/D layout: VGPR j holds M=j (lo half) / j+8
        hrow[row * HSTRIDE + 16 * t + c] = tanhf(acc[t][j]);
      }
    }
    asm volatile("s_wait_dscnt 0" ::: "memory");

    // ---- GEMM2: E = H @ W2 + b2 (A operand re-read from LDS in A layout) ---
    v8f acc2[4];
#pragma unroll
    for (int t = 0; t < 4; ++t) {
      v8f z;
#pragma unroll
      for (int j = 0; j < 8; ++j) z[j] = bias2[t];
      acc2[t] = z;
    }
#pragma unroll
    for (int kc = 0; kc < 16; ++kc) {
      v2f a = *(const v2f*)(hrow + c * HSTRIDE + 4 * kc + 2 * half);
#pragma unroll
      for (int t = 0; t < 4; ++t) {
        acc2[t] = __builtin_amdgcn_wmma_f32_16x16x4_f32(
            false, a, false, w2s[t][kc][lane], (short)0, acc2[t],
            false, false);
      }
    }

    // ---- per-edge gather * edge_emb, attention, atomic scatter -------------
#pragma unroll
    for (int j = 0; j < 8; ++j) {
      int erow = j + 8 * half;          // edge-in-tile this register row holds
      int dn = ib[erow];                // dest node
      int sn = ib[16 + erow];           // src node
      const float* srow = atom + (size_t)sn * HID;
      float m[4];
      float part = 0.0f;
#pragma unroll
      for (int t = 0; t < 4; ++t) {
        m[t] = acc2[t][j] * srow[16 * t + c];   // msg element (erow, 16t+c)
        part += m[t] * cf[t];
      }
      // row dot-product: reduce over the 16 lanes of this half
#pragma unroll
      for (int off = 1; off < 16; off <<= 1)
        part += __shfl_xor(part, off, 16);
      float attn = __expf(part);

      float* nrow = num + (size_t)dn * HID;
#pragma unroll
      for (int t = 0; t < 4; ++t)
        atomicAdd(nrow + 16 * t + c, m[t] * attn);   // L2-resident f32 atomics
      if (c == 0)
        atomicAdd(den + dn, attn);
    }
  }
}

// ---------------------------------------------------------------------------
// Epilogue: out = den > 0 ? num/den : num   (num already lives in d_out)
// ---------------------------------------------------------------------------
__global__ __launch_bounds__(256) void cfc_finalize_kernel(float* __restrict__ num,
                                                           const float* __restrict__ den) {
  int stride = gridDim.x * blockDim.x;
  for (int i = blockIdx.x * blockDim.x + threadIdx.x; i < NNODES * HID; i += stride) {
    float d = den[i >> 6];
    float v = num[i];
    num[i] = (d > 0.0f) ? (v / d) : v;
  }
}

extern "C" void kernel_launch(void* const* d_in, const int* in_sizes, int n_in,
                              void* d_out, int out_size, void* d_ws, size_t ws_size,
                              hipStream_t stream) {
  const float* atom = (const float*)d_in[0];  // hidden_atom_emb [50000,64]
  const float* eemb = (const float*)d_in[1];  // edge_emb        [800000,64]
  const float* W1   = (const float*)d_in[2];  // [64,64]
  const float* b1   = (const float*)d_in[3];  // [64]
  const float* W2   = (const float*)d_in[4];  // [64,64]
  const float* b2   = (const float*)d_in[5];  // [64]
  const float* coef = (const float*)d_in[6];  // [64]
  const int*   el   = (const int*)d_in[7];    // [800000,2]
  float* num = (float*)d_out;                 // [50000,64]
  float* den = (float*)d_ws;                  // [50000] scratch

  cfc_zero_kernel<<<512, 256, 0, stream>>>(num, den);
  cfc_edge_kernel<<<2500, WPB * 32, 0, stream>>>(atom, eemb, W1, b1, W2, b2,
                                                 coef, el, num, den);
  cfc_finalize_kernel<<<512, 256, 0, stream>>>(num, den);
}